// SyntacticGCN_39805756900150
// MI455X (gfx1250) — compile-verified
//
#include <hip/hip_runtime.h>
#include <math.h>
#include <cstdint>

#define NNODES 100000
#define NEDGES 600000
#define DM     128
#define LDSW   132   // padded LDS row stride (dwords): 132 mod 64 = 4 -> conflict-free

typedef __attribute__((ext_vector_type(2))) float v2f;
typedef __attribute__((ext_vector_type(4))) float v4f;
typedef __attribute__((ext_vector_type(8))) float v8f;

static __device__ __forceinline__ void wait_asynccnt0() {
#if __has_builtin(__builtin_amdgcn_s_wait_asynccnt)
    __builtin_amdgcn_s_wait_asynccnt(0);
#else
    asm volatile("s_wait_asynccnt 0x0" ::: "memory");
#endif
}

// ---------------------------------------------------------------------------
// Kernel 1: T[t] = inp @ W_t^T for t in {V_in, V_out, W_self, W_norel}
// Block = 256 threads (8 waves) on a 16-node block of one type.
// A tile (16x128 f32, 8 KB) is staged ONCE per block into LDS via the CDNA5
// async global->LDS copy (ASYNCcnt), then shared by all 8 waves; each wave
// computes one 16x16 f32 tile with V_WMMA_F32_16X16X4_F32 over K=128.
// T is streamed out with non-temporal stores (read once, later, randomly;
// 205 MB > 192 MB L2 -> keep L2 free for the edge phase's output atomics).
// ---------------------------------------------------------------------------
__global__ __launch_bounds__(256) void gemm4_wmma(
    const float* __restrict__ inp,
    const float* __restrict__ Vin, const float* __restrict__ Vout,
    const float* __restrict__ Wself, const float* __restrict__ Wnorel,
    float* __restrict__ Tws)
{
    __shared__ float As[16 * LDSW];

    const int tid   = threadIdx.x;
    const int lane  = tid & 31;
    const int wave  = tid >> 5;             // 0..7 -> 16-col tile
    const int node0 = blockIdx.x * 16;      // 100000/16 = 6250, exact
    const int type  = blockIdx.y;           // 0..3
    const float* W = (type == 0) ? Vin : (type == 1) ? Vout
                   : (type == 2) ? Wself : Wnorel;

    // ---- async stage A tile: 16 rows x 128 f32 = 512 x 16B chunks ----
#pragma unroll
    for (int i = 0; i < 2; ++i) {
        const int c   = tid * 2 + i;        // 0..511
        const int row = c >> 5;             // 0..15
        const int c16 = c & 31;             // 16B chunk within row
        const float* g = inp + (size_t)(node0 + row) * DM + c16 * 4;
        const uint32_t ldsoff =
            (uint32_t)(uintptr_t)(As + row * LDSW + c16 * 4);
        asm volatile("global_load_async_to_lds_b128 %0, %1, off"
                     :: "v"(ldsoff), "v"(g) : "memory");
    }
    wait_asynccnt0();
    __syncthreads();

    const int col0 = wave * 16;
    const int hl   = lane >> 4;             // K-pair select (ISA 16x4 layout)
    const int mn   = lane & 15;             // M index (A) / N index (B)

    // A from LDS (ds_load_b64, bank-conflict-free with stride 132);
    // B rows (= output columns) from global, hot in WGP$/L2 (64 KB/type).
    const float* aptr = As + mn * LDSW + 2 * hl;
    const float* brow = W + (size_t)(col0 + mn) * DM + 2 * hl;

    v8f c = {};
#pragma unroll
    for (int kk = 0; kk < DM; kk += 4) {
        v2f a = *(const v2f*)(aptr + kk);
        v2f b = *(const v2f*)(brow + kk);
        c = __builtin_amdgcn_wmma_f32_16x16x4_f32(
                false, a, false, b, (short)0, c, false, false);
    }

    // C/D layout: VGPR j -> row (j + 8*hl), col mn. Non-temporal stream-out.
    float* dst = Tws + (size_t)type * NNODES * DM
                     + (size_t)node0 * DM + col0;
#pragma unroll
    for (int j = 0; j < 8; ++j)
        __builtin_nontemporal_store(c[j], &dst[(size_t)(j + 8 * hl) * DM + mn]);
}

// ---------------------------------------------------------------------------
// Kernel 2: 4 gate pre-activations per node. One wave/node, wave32 shuffles.
// ---------------------------------------------------------------------------
__global__ __launch_bounds__(256) void gates_kernel(
    const float* __restrict__ inp,
    const float* __restrict__ gin, const float* __restrict__ gout,
    const float* __restrict__ gself, const float* __restrict__ gnorel,
    float* __restrict__ Gws)
{
    const int lane = threadIdx.x & 31;
    const int wave = threadIdx.x >> 5;
    const int node = blockIdx.x * 8 + wave;
    if (node >= NNODES) return;

    const float* row = inp + (size_t)node * DM;
    float s0 = 0.f, s1 = 0.f, s2 = 0.f, s3 = 0.f;
#pragma unroll
    for (int i = 0; i < 4; ++i) {
        const int k = lane + 32 * i;
        const float x = row[k];
        s0 += x * gin[k];   s1 += x * gout[k];
        s2 += x * gself[k]; s3 += x * gnorel[k];
    }
#pragma unroll
    for (int off = 16; off >= 1; off >>= 1) {
        s0 += __shfl_xor(s0, off, 32);
        s1 += __shfl_xor(s1, off, 32);
        s2 += __shfl_xor(s2, off, 32);
        s3 += __shfl_xor(s3, off, 32);
    }
    if (lane == 0) {
        Gws[0 * NNODES + node] = s0;
        Gws[1 * NNODES + node] = s1;
        Gws[2 * NNODES + node] = s2;
        Gws[3 * NNODES + node] = s3;
    }
}

// ---------------------------------------------------------------------------
// Kernel 3: per-edge gather + gate + scatter-add. One wave/edge, 4 f32/lane.
// T gather is non-temporal (one-shot 512 B rows, reuse distance >> L2) so the
// 51.2 MB output stays resident in the 192 MB L2 and the f32 atomics resolve
// on-chip. Bias tables / gates are tiny and stay regular-temporal.
// ---------------------------------------------------------------------------
__global__ __launch_bounds__(256) void edge_kernel(
    const int* __restrict__ deprel, const int* __restrict__ deparc,
    const int* __restrict__ eidx,
    const float* __restrict__ b_in, const float* __restrict__ b_out,
    const float* __restrict__ bg_in, const float* __restrict__ bg_out,
    const float* __restrict__ Tws, const float* __restrict__ Gws,
    float* __restrict__ out)
{
    const int lane = threadIdx.x & 31;
    const int wave = threadIdx.x >> 5;
    const int e = blockIdx.x * 8 + wave;    // NEDGES divisible by 8
    if (e >= NEDGES) return;

    const int t = deparc[e];
    const int r = deprel[e];
    const int s = eidx[e];
    const int d = eidx[NEDGES + e];

    float gpre = Gws[(size_t)t * NNODES + s];
    if (t == 0)      gpre += bg_in[r];
    else if (t == 1) gpre += bg_out[r];
    const float gate = 1.0f / (1.0f + __expf(-gpre));

    v4f v = __builtin_nontemporal_load(
        (const v4f*)(Tws + ((size_t)t * NNODES + s) * DM) + lane);
    if (t == 0) {
        const v4f b = *((const v4f*)(b_in + (size_t)r * DM) + lane);
        v += b;
    } else if (t == 1) {
        const v4f b = *((const v4f*)(b_out + (size_t)r * DM) + lane);
        v += b;
    }

    float* o = out + (size_t)d * DM + lane * 4;
    atomicAdd(o + 0, v.x * gate);
    atomicAdd(o + 1, v.y * gate);
    atomicAdd(o + 2, v.z * gate);
    atomicAdd(o + 3, v.w * gate);
}

// ---------------------------------------------------------------------------
extern "C" void kernel_launch(void* const* d_in, const int* in_sizes, int n_in,
                              void* d_out, int out_size, void* d_ws, size_t ws_size,
                              hipStream_t stream)
{
    const float* inp     = (const float*)d_in[0];
    const int*   deprel  = (const int*)  d_in[1];
    const int*   deparc  = (const int*)  d_in[2];
    const int*   eidx    = (const int*)  d_in[3];
    const float* Vin     = (const float*)d_in[4];
    const float* b_in    = (const float*)d_in[5];
    const float* gin     = (const float*)d_in[6];
    const float* bg_in   = (const float*)d_in[7];
    const float* Vout    = (const float*)d_in[8];
    const float* b_out   = (const float*)d_in[9];
    const float* gout    = (const float*)d_in[10];
    const float* bg_out  = (const float*)d_in[11];
    const float* Wself   = (const float*)d_in[12];
    const float* gself   = (const float*)d_in[13];
    const float* Wnorel  = (const float*)d_in[14];
    const float* gnorel  = (const float*)d_in[15];

    float* Tws = (float*)d_ws;                              // [4,N,D] = 204.8 MB
    float* Gws = Tws + (size_t)4 * NNODES * DM;             // [4,N]   = 1.6 MB

    hipMemsetAsync(d_out, 0, (size_t)NNODES * DM * sizeof(float), stream);

    gemm4_wmma<<<dim3(NNODES / 16, 4), 256, 0, stream>>>(
        inp, Vin, Vout, Wself, Wnorel, Tws);

    gates_kernel<<<(NNODES + 7) / 8, 256, 0, stream>>>(
        inp, gin, gout, gself, gnorel, Gws);

    edge_kernel<<<NEDGES / 8, 256, 0, stream>>>(
        deprel, deparc, eidx, b_in, b_out, bg_in, bg_out,
        Tws, Gws, (float*)d_out);
}